// CALayer_31447750541610
// MI455X (gfx1250) — compile-verified
//
#include <hip/hip_runtime.h>

typedef __attribute__((ext_vector_type(2))) float    v2f;
typedef __attribute__((ext_vector_type(8))) float    v8f;
typedef __attribute__((ext_vector_type(4))) unsigned v4u;
typedef __attribute__((ext_vector_type(8))) int      v8i;
typedef __attribute__((ext_vector_type(4))) int      v4i;

#define BATCH 16
#define CH 64
#define HW 65536              // 256*256
#define REDN 16
#define SPLITK 32
#define KSLICE (HW / SPLITK)  // 2048 columns per workgroup
#define KC 64                 // TDM chunk width (floats per row)
#define LDS_STRIDE 68         // 64 + 4 pad (TDM pad_amount=4 DWORDs) -> conflict-free
#define NCHUNK (KSLICE / KC)  // 32

// ---------------------------------------------------------------------------
// TDM: DMA one 64-row x 64-float fp32 tile (row stride HW floats) into LDS,
// inserting 4 DWORDs of padding after every 64 DWORDs stored (stride 68).
// D# per CDNA5 ISA ch.8; 2-D tensor -> groups 2/3 zero.
// ---------------------------------------------------------------------------
__device__ __forceinline__ void tdm_load_chunk(const float* gsrc, unsigned lds_byte) {
  unsigned long long A = (unsigned long long)(uintptr_t)gsrc;
  v4u g0 = { 1u,                                    // count=1, user descriptor
             lds_byte,                              // lds_addr
             (unsigned)A,                           // global_addr[31:0]
             ((unsigned)(A >> 32) & 0x01FFFFFFu) | (2u << 30) }; // [56:32] | type=2
  v8i g1 = { (int)((2u << 16)   // data_size = 4 bytes
                 | (1u << 20)   // pad_enable
                 | (5u << 22)   // pad_interval: 1<<5 = 32 x 8B = 64 DWORDs
                 | (3u << 25)), // pad_amount: 3+1 = 4 DWORDs
             0,                                       // atomic barrier addr = 0
             (int)((HW >> 16) | ((unsigned)CH << 16)),// tensor_dim0[31:16]=1 | dim1 lo
             (int)((unsigned)KC << 16),               // dim1 hi=0 | tile_dim0=64
             (int)CH,                                  // tile_dim1=64 | tile_dim2=0
             (int)HW,                                  // tensor_dim0_stride[31:0]
             0, 0 };                                   // stride hi / dim1_stride = 0
  v4i gz = { 0, 0, 0, 0 };
#if defined(__clang_major__) && (__clang_major__ >= 23)
  v8i gz8 = { 0, 0, 0, 0, 0, 0, 0, 0 };
  __builtin_amdgcn_tensor_load_to_lds(g0, g1, gz, gz, gz8, 0);
#else
  __builtin_amdgcn_tensor_load_to_lds(g0, g1, gz, gz, 0);
#endif
}

// ---------------------------------------------------------------------------
// Kernel 1: per-batch Gram matrix (X X^T raw sums) + per-channel pool sums.
// grid = BATCH * SPLITK blocks, 128 threads (4 waves). Double-buffered LDS:
// wave 0 issues TDM for chunk n+1 while all waves run WMMAs on chunk n.
// ---------------------------------------------------------------------------
__global__ __launch_bounds__(128) void gram_pool_kernel(
    const float* __restrict__ x, float* __restrict__ corr, float* __restrict__ pool) {
  __shared__ float buf[2][CH][LDS_STRIDE];
  __shared__ float pool_part[128];

  const int t     = threadIdx.x;
  const int bidx  = blockIdx.x / SPLITK;
  const int slice = blockIdx.x % SPLITK;
  const int kbase = slice * KSLICE;

  const float* xb = x + (size_t)bidx * CH * HW;

  const int wave = t >> 5;
  const int lane = t & 31;
  const int m    = lane & 15;
  const int hf   = lane >> 4;
  const int iblk = wave * 16;

  const v8f vzero = {0.f, 0.f, 0.f, 0.f, 0.f, 0.f, 0.f, 0.f};
  v8f acc[4];
  acc[0] = vzero; acc[1] = vzero; acc[2] = vzero; acc[3] = vzero;

  const int prow  = t & 63;
  const int phalf = t >> 6;
  float psum = 0.f;

  const unsigned lds0 = (unsigned)(uintptr_t)&buf[0][0][0];
  const unsigned lds1 = (unsigned)(uintptr_t)&buf[1][0][0];

  // Prologue: DMA chunk 0 into buffer 0.
  if (wave == 0) tdm_load_chunk(xb + kbase, lds0);

  for (int chunk = 0; chunk < NCHUNK; ++chunk) {
    const int cur = chunk & 1;
    if (wave == 0) __builtin_amdgcn_s_wait_tensorcnt(0);  // chunk's data in LDS
    __syncthreads();  // publish DMA data; all waves done reading buf[1-cur]
    if (wave == 0 && chunk + 1 < NCHUNK)
      tdm_load_chunk(xb + kbase + (chunk + 1) * KC, cur ? lds0 : lds1);

    // Pool partial sums from LDS.
    #pragma unroll 8
    for (int cc = 0; cc < KC / 2; ++cc)
      psum += buf[cur][prow][phalf * (KC / 2) + cc];

    // WMMA: D(16x16) += A(16x4) x B(4x16). A/B fragment layouts coincide for
    // a Gram matrix: lane(m,hf) reads row blk+m, cols kk+2*hf .. +1.
    for (int kk = 0; kk < KC; kk += 4) {
      v2f a = *reinterpret_cast<const v2f*>(&buf[cur][iblk + m][kk + 2 * hf]);
      #pragma unroll
      for (int j = 0; j < 4; ++j) {
        v2f bf = *reinterpret_cast<const v2f*>(&buf[cur][j * 16 + m][kk + 2 * hf]);
        acc[j] = __builtin_amdgcn_wmma_f32_16x16x4_f32(
            false, a, false, bf, (short)0, acc[j], false, false);
      }
    }
  }

  // Reduce + commit pool partials.
  pool_part[t] = psum;
  __syncthreads();
  if (t < 64)
    unsafeAtomicAdd(&pool[bidx * CH + t], pool_part[t] + pool_part[t + 64]);

  // Commit Gram tiles. C/D layout: VGPR r on lane(m,hf) = D[r + 8*hf][m].
  float* cb = corr + (size_t)bidx * CH * CH;
  #pragma unroll
  for (int j = 0; j < 4; ++j) {
    #pragma unroll
    for (int r = 0; r < 8; ++r) {
      int row = iblk + r + 8 * hf;
      int col = j * 16 + m;
      unsafeAtomicAdd(&cb[row * CH + col], acc[j][r]);
    }
  }
}

// ---------------------------------------------------------------------------
// Kernel 2: gating MLP on [B,C] — tiny, one 1024-thread block.
// ---------------------------------------------------------------------------
__global__ __launch_bounds__(1024) void gate_kernel(
    const float* __restrict__ corr, const float* __restrict__ pool,
    const float* __restrict__ Wrow, const float* __restrict__ brow,
    const float* __restrict__ W1, const float* __restrict__ b1,
    const float* __restrict__ W2, const float* __restrict__ b2,
    float* __restrict__ g) {
  __shared__ float ybuf[BATCH][CH];
  __shared__ float zbuf[BATCH][REDN];
  const int t  = threadIdx.x;
  const int bb = t >> 6;
  const int c  = t & 63;
  const float invn = 1.0f / (float)HW;

  const float* crow = corr + ((size_t)bb * CH + c) * CH;
  const float* wr   = Wrow + c * CH;
  float s = 0.f;
  #pragma unroll 8
  for (int d = 0; d < CH; ++d) s += crow[d] * wr[d];
  ybuf[bb][c] = pool[bb * CH + c] * invn + s * invn + brow[c];
  __syncthreads();

  if (t < BATCH * REDN) {
    int b_i = t >> 4, r = t & 15;
    float a = b1[r];
    #pragma unroll 8
    for (int d = 0; d < CH; ++d) a += ybuf[b_i][d] * W1[r * CH + d];
    zbuf[b_i][r] = a > 0.f ? a : 0.f;
  }
  __syncthreads();

  float a = b2[c];
  #pragma unroll
  for (int d = 0; d < REDN; ++d) a += zbuf[bb][d] * W2[c * REDN + d];
  g[t] = 1.0f / (1.0f + __expf(-a));
}

// ---------------------------------------------------------------------------
// Kernel 3: out = x * g[b,c], float4-vectorized stream (512 MB traffic).
// ---------------------------------------------------------------------------
__global__ __launch_bounds__(256) void scale_kernel(
    const float* __restrict__ x, const float* __restrict__ g,
    float* __restrict__ out) {
  size_t i = (size_t)blockIdx.x * 256 + threadIdx.x;
  int bc = (int)(i >> 14);  // HW/4 = 16384 float4 per (b,c) plane
  float gv = g[bc];
  float4 v = reinterpret_cast<const float4*>(x)[i];
  v.x *= gv; v.y *= gv; v.z *= gv; v.w *= gv;
  reinterpret_cast<float4*>(out)[i] = v;
}

// ---------------------------------------------------------------------------
extern "C" void kernel_launch(void* const* d_in, const int* in_sizes, int n_in,
                              void* d_out, int out_size, void* d_ws, size_t ws_size,
                              hipStream_t stream) {
  const float* x    = (const float*)d_in[0];
  const float* Wrow = (const float*)d_in[1];
  const float* brow = (const float*)d_in[2];
  const float* W1   = (const float*)d_in[3];
  const float* b1   = (const float*)d_in[4];
  const float* W2   = (const float*)d_in[5];
  const float* b2   = (const float*)d_in[6];
  float* out = (float*)d_out;

  float* corr = (float*)d_ws;               // B*C*C = 65536 floats
  float* pool = corr + BATCH * CH * CH;     // B*C   = 1024 floats
  float* g    = pool + BATCH * CH;          // B*C   = 1024 floats

  hipMemsetAsync(d_ws, 0, (size_t)(BATCH * CH * CH + BATCH * CH) * sizeof(float),
                 stream);

  gram_pool_kernel<<<BATCH * SPLITK, 128, 0, stream>>>(x, corr, pool);
  gate_kernel<<<1, BATCH * CH, 0, stream>>>(corr, pool, Wrow, brow, W1, b1, W2, b2, g);
  scale_kernel<<<(BATCH * CH * HW / 4) / 256, 256, 0, stream>>>(x, g, out);
}